// PeriodicGeometryAdapter_85667417686321
// MI455X (gfx1250) — compile-verified
//
#include <hip/hip_runtime.h>
#include <hip/hip_bf16.h>
#include <math.h>

typedef __attribute__((ext_vector_type(2))) float v2f;
typedef __attribute__((ext_vector_type(8))) float v8f;

#define BSZ  8
#define NPTS 256
#define NOFF 27

// ---- workspace layout (float offsets) ----
#define WS_U    0                              // B*N*3   u = G x
#define WS_Q    (BSZ*NPTS*3)                   // B*N     q = x.Gx
#define WS_AM   (WS_Q  + BSZ*NPTS)             // B*27*N  a_mk = 2 x.(G o_k)
#define WS_AP   (WS_AM + BSZ*NOFF*NPTS)        // B*27*N  a'_nk = a_nk + o.Go
#define WS_CELL (WS_AP + BSZ*NOFF*NPTS)        // B

// ---- output layout (float offsets, reference tuple order) ----
#define O_PDM  0                               // B*N*N*3 pair_delta_min
#define O_DIST (BSZ*NPTS*NPTS*3)               // B*N*N   pair_dist
#define O_D2   (O_DIST + BSZ*NPTS*NPTS)        // B*N*N   pair_dist2 (also GEMM scratch)
#define O_DN   (O_D2   + BSZ*NPTS*NPTS)        // B*N*N   pair_dist_norm
#define O_MASK (O_DN   + BSZ*NPTS*NPTS)        // B*N*N   pair_mask (as 0/1 float)
#define O_LAT  (O_MASK + BSZ*NPTS*NPTS)        // B*6     lattice_y1
#define O_GRAM (O_LAT  + BSZ*6)                // B*9     gram
#define O_CELL (O_GRAM + BSZ*9)                // B       cell_scale

// ---------------------------------------------------------------------------
// Kernel 1: per-batch Gram tensor + affine precomputes
// ---------------------------------------------------------------------------
__global__ __launch_bounds__(256) void pg_setup(const float* __restrict__ coords,
                                                const float* __restrict__ lattice,
                                                float* __restrict__ out,
                                                float* __restrict__ ws) {
    const int b   = blockIdx.x;
    const int tid = threadIdx.x;
    __shared__ float g[9];
    __shared__ float wk[NOFF][3];
    __shared__ float ck[NOFF];

    if (tid == 0) {
        const float* lat = lattice + b * 6;
        float la = expf(lat[0]), lb = expf(lat[1]), lc = expf(lat[2]);
        float ca = tanhf(lat[3]), cb = tanhf(lat[4]), cg = tanhf(lat[5]);
        g[0] = la * la;      g[1] = la * lb * cg; g[2] = la * lc * cb;
        g[3] = g[1];         g[4] = lb * lb;      g[5] = lb * lc * ca;
        g[6] = g[2];         g[7] = g[5];         g[8] = lc * lc;
        float cell = fmaxf((la + lb + lc) * (1.0f / 3.0f), 1e-8f);
        ws[WS_CELL + b] = cell;
        out[O_CELL + b] = cell;
        for (int i = 0; i < 6; ++i) out[O_LAT + b * 6 + i] = lat[i];
        for (int i = 0; i < 9; ++i) out[O_GRAM + b * 9 + i] = g[i];
    }
    __syncthreads();

    if (tid < NOFF) {
        float ox = (float)(tid / 9) - 1.0f;
        float oy = (float)((tid / 3) % 3) - 1.0f;
        float oz = (float)(tid % 3) - 1.0f;
        float w0 = g[0] * ox + g[1] * oy + g[2] * oz;
        float w1 = g[3] * ox + g[4] * oy + g[5] * oz;
        float w2 = g[6] * ox + g[7] * oy + g[8] * oz;
        wk[tid][0] = w0; wk[tid][1] = w1; wk[tid][2] = w2;
        ck[tid] = ox * w0 + oy * w1 + oz * w2;
    }
    __syncthreads();

    const int n = tid;
    float x0 = coords[(b * NPTS + n) * 3 + 0];
    float x1 = coords[(b * NPTS + n) * 3 + 1];
    float x2 = coords[(b * NPTS + n) * 3 + 2];
    float u0 = g[0] * x0 + g[1] * x1 + g[2] * x2;
    float u1 = g[3] * x0 + g[4] * x1 + g[5] * x2;
    float u2 = g[6] * x0 + g[7] * x1 + g[8] * x2;
    ws[WS_U + (b * NPTS + n) * 3 + 0] = u0;
    ws[WS_U + (b * NPTS + n) * 3 + 1] = u1;
    ws[WS_U + (b * NPTS + n) * 3 + 2] = u2;
    ws[WS_Q + b * NPTS + n] = x0 * u0 + x1 * u1 + x2 * u2;

    #pragma unroll
    for (int k = 0; k < NOFF; ++k) {
        float a2 = 2.0f * (x0 * wk[k][0] + x1 * wk[k][1] + x2 * wk[k][2]);
        ws[WS_AM + (b * NOFF + k) * NPTS + n] = a2;
        ws[WS_AP + (b * NOFF + k) * NPTS + n] = a2 + ck[k];
    }
}

// ---------------------------------------------------------------------------
// Kernel 2: base(n,m) = q_n + q_m - 2 * (X . (G X)^T)  via V_WMMA_F32_16X16X4_F32
// One wave per 16x16 tile. A = 16x4 coords (K=3 padded), B = 4x16 u columns.
// ---------------------------------------------------------------------------
__global__ __launch_bounds__(32) void pg_gemm_base(const float* __restrict__ coords,
                                                   const float* __restrict__ ws,
                                                   float* __restrict__ out) {
    const int lane = threadIdx.x;
    const int half = lane >> 4;      // 0: K=0,1   1: K=2,3(pad)
    const int lm   = lane & 15;
    const int ti = blockIdx.x, tj = blockIdx.y, b = blockIdx.z;
    const int row = ti * 16 + lm;    // A-matrix row (M)
    const int col = tj * 16 + lm;    // B-matrix column (N)

    const float* xr = coords + (b * NPTS + row) * 3;
    const float* uc = ws + WS_U + (b * NPTS + col) * 3;

    v2f A, Bm;
    A.x  = half ? xr[2] : xr[0];
    A.y  = half ? 0.0f  : xr[1];
    Bm.x = half ? uc[2] : uc[0];
    Bm.y = half ? 0.0f  : uc[1];

    v8f C = {};
    v8f D = __builtin_amdgcn_wmma_f32_16x16x4_f32(
        /*neg_a=*/false, A, /*neg_b=*/false, Bm,
        /*c_mod=*/(short)0, C, /*reuse_a=*/false, /*reuse_b=*/false);

    const float* q = ws + WS_Q + b * NPTS;
    const float qc = q[col];
    float* baseOut = out + O_D2 + (size_t)b * NPTS * NPTS;
    #pragma unroll
    for (int r = 0; r < 8; ++r) {
        int grow = ti * 16 + r + 8 * half;   // D layout: row = r + 8*(lane>=16)
        baseOut[(size_t)grow * NPTS + col] = q[grow] + qc - 2.0f * D[r];
    }
}

// ---------------------------------------------------------------------------
// Kernel 3: 27-way argmin + output assembly. One block = one (b, n) row, m = tid.
// ---------------------------------------------------------------------------
__global__ __launch_bounds__(256) void pg_argmin(const float* __restrict__ coords,
                                                 const unsigned char* __restrict__ pad,
                                                 const float* __restrict__ ws,
                                                 float* __restrict__ out) {
    const int b = blockIdx.x >> 8;
    const int n = blockIdx.x & 255;
    const int m = threadIdx.x;

    __shared__ float sAM[NOFF * NPTS];   // a_mk for the whole row's batch, [k][m]
    __shared__ float sAP[NOFF];          // a'_nk for this n (block-uniform)
    __shared__ float sXn[3];
    __shared__ float sCell;
    __shared__ int   sPadN;

    const float* aMsrc = ws + WS_AM + b * NOFF * NPTS;
    for (int j = threadIdx.x; j < NOFF * NPTS; j += 256) sAM[j] = aMsrc[j];
    if (m < NOFF)            sAP[m]      = ws[WS_AP + (b * NOFF + m) * NPTS + n];
    if (m >= 32 && m < 35)   sXn[m - 32] = coords[(b * NPTS + n) * 3 + (m - 32)];
    if (m == 35)             sCell       = ws[WS_CELL + b];
    if (m == 36)             sPadN       = pad[b * NPTS + n];
    __syncthreads();

    const size_t pidx = ((size_t)b * NPTS + n) * NPTS + m;
    float base = out[O_D2 + pidx];       // written by pg_gemm_base every launch

    float best = 3.402823466e38f;
    int   bk   = 0;
    #pragma unroll
    for (int k = 0; k < NOFF; ++k) {
        float d2 = base + sAP[k] - sAM[k * NPTS + m];
        bool lt = d2 < best;             // strict '<' => first minimum, like argmin
        best = lt ? d2 : best;
        bk   = lt ? k  : bk;
    }

    float xm0 = coords[(b * NPTS + m) * 3 + 0];
    float xm1 = coords[(b * NPTS + m) * 3 + 1];
    float xm2 = coords[(b * NPTS + m) * 3 + 2];
    float dx = sXn[0] - xm0, dy = sXn[1] - xm1, dz = sXn[2] - xm2;

    float ox = (float)(bk / 9) - 1.0f;
    float oy = (float)((bk / 3) % 3) - 1.0f;
    float oz = (float)(bk % 3) - 1.0f;

    bool  valid = (sPadN == 0) && (pad[b * NPTS + m] == 0);
    float fm    = valid ? 1.0f : 0.0f;

    float d2c  = fmaxf(best, 0.0f);
    float dist = sqrtf(d2c);

    out[O_PDM + pidx * 3 + 0] = valid ? (dx + ox) : 0.0f;
    out[O_PDM + pidx * 3 + 1] = valid ? (dy + oy) : 0.0f;
    out[O_PDM + pidx * 3 + 2] = valid ? (dz + oz) : 0.0f;
    out[O_DIST + pidx] = valid ? dist : 0.0f;
    out[O_D2   + pidx] = valid ? d2c  : 0.0f;
    out[O_DN   + pidx] = valid ? (dist / sCell) : 0.0f;
    out[O_MASK + pidx] = fm;
}

// ---------------------------------------------------------------------------
extern "C" void kernel_launch(void* const* d_in, const int* in_sizes, int n_in,
                              void* d_out, int out_size, void* d_ws, size_t ws_size,
                              hipStream_t stream) {
    (void)in_sizes; (void)n_in; (void)out_size; (void)ws_size;
    const float*         coords  = (const float*)d_in[0];
    const unsigned char* padmask = (const unsigned char*)d_in[1];
    const float*         lattice = (const float*)d_in[2];
    float* out = (float*)d_out;
    float* ws  = (float*)d_ws;

    pg_setup<<<BSZ, 256, 0, stream>>>(coords, lattice, out, ws);
    pg_gemm_base<<<dim3(NPTS / 16, NPTS / 16, BSZ), 32, 0, stream>>>(coords, ws, out);
    pg_argmin<<<BSZ * NPTS, 256, 0, stream>>>(coords, padmask, ws, out);
}